// TASI_GNN_57724360458999
// MI455X (gfx1250) — compile-verified
//
#include <hip/hip_runtime.h>
#include <math.h>

// Problem constants (match the reference)
#define BB 512
#define LL 50
#define EE 128
#define VV 100000
#define DD 256      // D = 2E
#define L2 100      // 2L
#define EPSF 1e-12f

typedef float v2f __attribute__((ext_vector_type(2)));
typedef float v8f __attribute__((ext_vector_type(8)));

#define WMMA4(a, b, c) __builtin_amdgcn_wmma_f32_16x16x4_f32(false, (a), false, (b), (short)0, (c), false, false)

__device__ __forceinline__ int imin(int a, int b) { return a < b ? a : b; }

// ---------------------------------------------------------------------------
// Generic fp32 WMMA GEMM:  C = act( outScale * (A @ opB(B)) * colScale + bias )
//   TB=1 : Bm is (N,K) row-major  -> computes A @ Bm^T   (weight matrices)
//   TB=0 : Bm is (K,N) row-major  -> computes A @ Bm     (adjacency/attention)
// One wave computes a 16x64 tile (4 accumulators) via V_WMMA_F32_16X16X4_F32.
// Interior tiles: unguarded v2f loads + 1-deep software pipeline. Edge tiles:
// clamped addresses + multiply-by-mask zeroing. The mask multiply makes the
// loaded value unconditionally used, so the backend cannot sink the load
// under a divergent branch (no saveexec/cbranch in the inner loop) and EXEC
// stays all-ones through every WMMA.
// ---------------------------------------------------------------------------
template<int TB, int ACT>
__global__ void gemm_wmma_kernel(const float* __restrict__ A, const float* __restrict__ Bm,
                                 const float* __restrict__ bias, const float* __restrict__ colScale,
                                 float* __restrict__ C,
                                 int M, int N, int K, int lda, int ldb, int ldc,
                                 long sA, long sB, long sC, long sBias, float outScale)
{
    const int wave = threadIdx.x >> 5;
    const int lane = threadIdx.x & 31;
    const int ll   = lane & 15;
    const int lh   = lane >> 4;              // 0 or 1
    const int n0   = (blockIdx.x * 4 + wave) * 64;
    const int m0   = blockIdx.y * 16;
    if (n0 >= N || m0 >= M) return;          // wave-uniform exit
    const long bz = blockIdx.z;
    A  += bz * sA;
    Bm += bz * sB;
    C  += bz * sC;
    const float* brow = bias ? (bias + bz * sBias) : (const float*)0;

    v8f acc0 = {0.f,0.f,0.f,0.f,0.f,0.f,0.f,0.f};
    v8f acc1 = acc0, acc2 = acc0, acc3 = acc0;

    const bool fast = ((K & 3) == 0) && (m0 + 16 <= M) && (n0 + 64 <= N);

    if (fast) {
        // ---------------- interior tile: vector loads, pipelined ----------------
        const float* Ap = A + (long)(m0 + ll) * lda + 2 * lh;
        v2f a  = *(const v2f*)Ap;
        v2f b0, b1, b2, b3;
        if (TB) {
            const float* Bp = Bm + (long)(n0 + ll) * ldb + 2 * lh;
            b0 = *(const v2f*)(Bp);
            b1 = *(const v2f*)(Bp + (long)16 * ldb);
            b2 = *(const v2f*)(Bp + (long)32 * ldb);
            b3 = *(const v2f*)(Bp + (long)48 * ldb);
            for (int k0 = 4; k0 < K; k0 += 4) {
                v2f an = *(const v2f*)(Ap + k0);
                v2f c0 = *(const v2f*)(Bp + k0);
                v2f c1 = *(const v2f*)(Bp + k0 + (long)16 * ldb);
                v2f c2 = *(const v2f*)(Bp + k0 + (long)32 * ldb);
                v2f c3 = *(const v2f*)(Bp + k0 + (long)48 * ldb);
                acc0 = WMMA4(a, b0, acc0);
                acc1 = WMMA4(a, b1, acc1);
                acc2 = WMMA4(a, b2, acc2);
                acc3 = WMMA4(a, b3, acc3);
                a = an; b0 = c0; b1 = c1; b2 = c2; b3 = c3;
            }
        } else {
            const float* Bp = Bm + (long)(2 * lh) * ldb + (n0 + ll);
            auto ldB = [&](int k0, int j) {
                const float* p = Bp + (long)k0 * ldb + 16 * j;
                v2f r; r.x = p[0]; r.y = p[ldb]; return r;
            };
            b0 = ldB(0, 0); b1 = ldB(0, 1); b2 = ldB(0, 2); b3 = ldB(0, 3);
            for (int k0 = 4; k0 < K; k0 += 4) {
                v2f an = *(const v2f*)(Ap + k0);
                v2f c0 = ldB(k0, 0);
                v2f c1 = ldB(k0, 1);
                v2f c2 = ldB(k0, 2);
                v2f c3 = ldB(k0, 3);
                acc0 = WMMA4(a, b0, acc0);
                acc1 = WMMA4(a, b1, acc1);
                acc2 = WMMA4(a, b2, acc2);
                acc3 = WMMA4(a, b3, acc3);
                a = an; b0 = c0; b1 = c1; b2 = c2; b3 = c3;
            }
        }
        acc0 = WMMA4(a, b0, acc0);
        acc1 = WMMA4(a, b1, acc1);
        acc2 = WMMA4(a, b2, acc2);
        acc3 = WMMA4(a, b3, acc3);
    } else {
        // -------- edge tile: clamped loads, zeroed via mask multiplication ------
        const int ar   = m0 + ll;
        const int arc  = imin(ar, M - 1);
        const float mRow = (ar < M) ? 1.f : 0.f;
        const float* Arow = A + (long)arc * lda;
        int   bcc[4];
        float mCol[4];
#pragma unroll
        for (int j = 0; j < 4; ++j) {
            const int bc = n0 + 16 * j + ll;
            bcc[j]  = imin(bc, N - 1);
            mCol[j] = (bc < N) ? 1.f : 0.f;
        }
        v8f acc[4] = {acc0, acc1, acc2, acc3};
        for (int k0 = 0; k0 < K; k0 += 4) {
            const int kk  = k0 + 2 * lh;
            const int kc0 = imin(kk, K - 1);
            const int kc1 = imin(kk + 1, K - 1);
            const float mk0 = (kk     < K) ? 1.f : 0.f;
            const float mk1 = (kk + 1 < K) ? 1.f : 0.f;
            v2f a;
            a.x = Arow[kc0] * (mRow * mk0);
            a.y = Arow[kc1] * (mRow * mk1);
#pragma unroll
            for (int j = 0; j < 4; ++j) {
                float v0, v1;
                if (TB) {
                    const float* Brow = Bm + (long)bcc[j] * ldb;
                    v0 = Brow[kc0]; v1 = Brow[kc1];
                } else {
                    v0 = Bm[(long)kc0 * ldb + bcc[j]];
                    v1 = Bm[(long)kc1 * ldb + bcc[j]];
                }
                v2f b;
                b.x = v0 * (mk0 * mCol[j]);
                b.y = v1 * (mk1 * mCol[j]);
                acc[j] = WMMA4(a, b, acc[j]);
            }
        }
        acc0 = acc[0]; acc1 = acc[1]; acc2 = acc[2]; acc3 = acc[3];
    }

    // ---------------- epilogue ----------------
    auto finish = [&](float v, int n) {
        v *= outScale;
        if (colScale) v *= colScale[n];
        if (brow)     v += brow[n];
        if (ACT == 1) v = fmaxf(v, 0.f);
        else if (ACT == 2) {                       // SELU
            const float sc = 1.0507009873554805f;
            const float al = 1.6732632423543772f;
            v = (v > 0.f) ? sc * v : sc * al * expm1f(v);
        }
        return v;
    };
#pragma unroll
    for (int r = 0; r < 8; ++r) {
        const int m = m0 + r + 8 * lh;             // C/D layout: reg r -> rows r, r+8
        const int n = n0 + ll;
        if (fast) {
            float* Crow = C + (long)m * ldc + n;
            Crow[0]  = finish(acc0[r], n);
            Crow[16] = finish(acc1[r], n + 16);
            Crow[32] = finish(acc2[r], n + 32);
            Crow[48] = finish(acc3[r], n + 48);
        } else if (m < M) {
            float* Crow = C + (long)m * ldc;
            if (n      < N) Crow[n     ] = finish(acc0[r], n);
            if (n + 16 < N) Crow[n + 16] = finish(acc1[r], n + 16);
            if (n + 32 < N) Crow[n + 32] = finish(acc2[r], n + 32);
            if (n + 48 < N) Crow[n + 48] = finish(acc3[r], n + 48);
        }
    }
}

static void launch_gemm(hipStream_t s, int tb, int act,
                        const float* A, const float* Bm, const float* bias,
                        const float* colScale, float* C,
                        int M, int N, int K, int lda, int ldb, int ldc,
                        long sA, long sB, long sC, long sBias,
                        float outScale, int batch)
{
    dim3 block(128, 1, 1);
    dim3 grid((N + 255) / 256, (M + 15) / 16, batch);   // wave = 16x64 tile
#define GW(TBv, ACTv) gemm_wmma_kernel<TBv, ACTv><<<grid, block, 0, s>>>( \
        A, Bm, bias, colScale, C, M, N, K, lda, ldb, ldc, sA, sB, sC, sBias, outScale)
    if (tb) { if (act == 0) GW(1,0); else if (act == 1) GW(1,1); else GW(1,2); }
    else    { if (act == 0) GW(0,0); else if (act == 1) GW(0,1); else GW(0,2); }
#undef GW
}

// --------------------------- block reductions ------------------------------
template<int BD>
__device__ __forceinline__ float breduce_sum(float v, float* red) {
    const int tid = threadIdx.x;
    red[tid] = v; __syncthreads();
#pragma unroll
    for (int st = BD / 2; st > 0; st >>= 1) {
        if (tid < st) red[tid] += red[tid + st];
        __syncthreads();
    }
    float r = red[0]; __syncthreads();
    return r;
}
template<int BD>
__device__ __forceinline__ float breduce_max(float v, float* red) {
    const int tid = threadIdx.x;
    red[tid] = v; __syncthreads();
#pragma unroll
    for (int st = BD / 2; st > 0; st >>= 1) {
        if (tid < st) red[tid] = fmaxf(red[tid], red[tid + st]);
        __syncthreads();
    }
    float r = red[0]; __syncthreads();
    return r;
}

// --------------------------- elementwise kernels ---------------------------
__global__ void build_x_kernel(const float* __restrict__ emb, const float* __restrict__ pos,
                               const int* __restrict__ item, float* __restrict__ X, int total)
{
    int i = blockIdx.x * blockDim.x + threadIdx.x;
    if (i >= total) return;
    int d = i % DD;
    int l = (i / DD) % LL;
    int b = i / (DD * LL);
    float v;
    if (d < EE) v = emb[(long)item[b * LL + l] * EE + d];
    else        v = pos[l * EE + (d - EE)];
    X[i] = v;
}

__global__ void gru_kernel(const float* __restrict__ GI, const float* __restrict__ GH,
                           const float* __restrict__ X, float* __restrict__ XN, int total)
{
    int i = blockIdx.x * blockDim.x + threadIdx.x;
    if (i >= total) return;
    int d = i % DD;
    long row = i / DD;
    const float* gi = GI + row * 3 * DD;
    const float* gh = GH + row * 3 * DD;
    float r = 1.f / (1.f + expf(-(gi[d] + gh[d])));
    float z = 1.f / (1.f + expf(-(gi[DD + d] + gh[DD + d])));
    float n = tanhf(gi[2 * DD + d] + r * gh[2 * DD + d]);
    XN[i] = (1.f - z) * n + z * X[i];
}

__global__ void gather_kernel(const float* __restrict__ XN, const int* __restrict__ alias,
                              float* __restrict__ XTP, int total)
{
    int i = blockIdx.x * blockDim.x + threadIdx.x;
    if (i >= total) return;
    int d = i % DD;
    int l = (i / DD) % L2;
    int b = i / (DD * L2);
    float v = 0.f;
    if (l < LL) v = XN[((long)b * LL + alias[b * LL + l]) * DD + d];
    XTP[i] = v;
}

__global__ void alpha1_kernel(const float* __restrict__ ba, float* __restrict__ A1)
{
    int i = blockIdx.x * blockDim.x + threadIdx.x;
    // X_tp[:, -1, :] is the zero-padded half -> alpha1 = sigmoid(ba) + 1 for all b
    if (i < BB) A1[i] = 1.f / (1.f + expf(-ba[0])) + 1.f;
}

// entmax bisection over one row per block (256 threads, d <= 256)
__global__ void entmax_kernel(const float* __restrict__ in, float* __restrict__ out,
                              int d, const float* __restrict__ alphaPtr, int rowDiv)
{
    __shared__ float red[256];
    const long row = blockIdx.x;
    const int tid = threadIdx.x;
    const float alpha = alphaPtr[row / rowDiv];
    const float am1 = alpha - 1.f;
    const float inv = 1.f / am1;

    float xs = -INFINITY;
    if (tid < d) xs = in[row * d + tid] * am1;

    float maxv = breduce_max<256>(xs, red);
    float tau_lo = maxv - 1.f;
    float tau_hi = maxv - powf(1.f / (float)d, am1);
    float dm = tau_hi - tau_lo;

    float z = xs - tau_lo;
    float pv = (z > 0.f) ? powf(z, inv) : 0.f;
    float f_lo = breduce_sum<256>(pv, red) - 1.f;

    for (int it = 0; it < 30; ++it) {
        dm *= 0.5f;
        float tau_m = tau_lo + dm;
        z = xs - tau_m;
        pv = (z > 0.f) ? powf(z, inv) : 0.f;
        float f_m = breduce_sum<256>(pv, red) - 1.f;
        if (f_m * f_lo >= 0.f) tau_lo = tau_m;   // uniform across block
    }
    z = xs - (tau_lo + dm);
    pv = (z > 0.f) ? powf(z, inv) : 0.f;
    float ssum = breduce_sum<256>(pv, red);
    if (tid < d) out[row * d + tid] = pv / (ssum + EPSF);
}

__global__ void ln_kernel(const float* __restrict__ F2, const float* __restrict__ R,
                          const float* __restrict__ g, const float* __restrict__ bt,
                          float* __restrict__ CH)
{
    __shared__ float red[256];
    const long row = blockIdx.x;
    const int tid = threadIdx.x;       // == d (blockDim 256)
    float v = F2[row * DD + tid] + R[row * DD + tid];
    float m = breduce_sum<256>(v, red) * (1.f / DD);
    float dv = v - m;
    float var = breduce_sum<256>(dv * dv, red) * (1.f / DD);
    CH[row * DD + tid] = dv * rsqrtf(var + EPSF) * g[tid] + bt[tid];
}

__global__ void clast_kernel(const float* __restrict__ CH, const int* __restrict__ mask,
                             float* __restrict__ CLAST, float* __restrict__ TGT,
                             float* __restrict__ TC)
{
    __shared__ int ssl;
    const int b = blockIdx.x;
    const int tid = threadIdx.x;       // 256
    if (tid == 0) {
        int s = 0;
        for (int l = 0; l < LL; ++l) s += mask[b * LL + l];
        ssl = s - 1;
    }
    __syncthreads();
    const int sl = ssl;
    float cl = CH[((long)b * L2 + sl) * DD + tid];
    float tg = CH[((long)b * L2 + (L2 - 1)) * DD + tid];
    CLAST[b * DD + tid] = cl;
    TGT[b * DD + tid]   = tg;
    TC[(long)b * 2 * DD + tid]      = tg;   // concat([target, C_last])
    TC[(long)b * 2 * DD + DD + tid] = cl;
}

__global__ void sim_kernel(const float* __restrict__ CLAST, const float* __restrict__ TGT,
                           float* __restrict__ SIMV)
{
    __shared__ float red[256];
    const int b = blockIdx.x;
    const int tid = threadIdx.x;
    float c = CLAST[b * DD + tid];
    float t = TGT[b * DD + tid];
    float dot = breduce_sum<256>(c * t, red);
    float nc  = breduce_sum<256>(c * c, red);
    float nt  = breduce_sum<256>(t * t, red);
    if (tid == 0) SIMV[b] = dot / ((sqrtf(nc) + EPSF) * (sqrtf(nt) + EPSF));
}

__global__ void thr_kernel(const float* __restrict__ SIMV, float* __restrict__ THR)
{
    __shared__ float red[512];
    const int tid = threadIdx.x;
    float s = breduce_sum<512>(SIMV[tid], red);
    if (tid == 0) THR[0] = 0.5f * s * (1.f / BB);   // GAMMA * mean(sim)
}

__global__ void select_kernel(const float* __restrict__ SIMV, const float* __restrict__ THR,
                              const float* __restrict__ POSV, float* __restrict__ TGT, int total)
{
    int i = blockIdx.x * blockDim.x + threadIdx.x;
    if (i >= total) return;
    int b = i / DD;
    if (SIMV[b] - THR[0] > 0.f) TGT[i] = POSV[i];
}

__global__ void alpha2_kernel(const float* __restrict__ TGT, const float* __restrict__ Wa,
                              const float* __restrict__ ba, float* __restrict__ A2)
{
    __shared__ float red[256];
    const int b = blockIdx.x;
    const int tid = threadIdx.x;
    float s = breduce_sum<256>(TGT[b * DD + tid] * Wa[tid], red);
    if (tid == 0) A2[b] = 1.f / (1.f + expf(-(s + ba[0]))) + 1.f;
}

__global__ void cg_kernel(const float* __restrict__ PG, const float* __restrict__ XTP,
                          const float* __restrict__ TGT, float* __restrict__ HCAT)
{
    const int b = blockIdx.x;
    const int d = threadIdx.x;         // 256
    float acc = 0.f;
    for (int l = 0; l < LL; ++l)
        acc += PG[((long)b * LL + l) * DD + d] * XTP[((long)b * L2 + l) * DD + d];
    HCAT[(long)b * 2 * DD + d]      = acc;            // C_global
    HCAT[(long)b * 2 * DD + DD + d] = TGT[b * DD + d];
}

__global__ void hn_kernel(const float* __restrict__ H, float* __restrict__ HN)
{
    __shared__ float red[128];
    const int b = blockIdx.x;
    const int tid = threadIdx.x;       // 128 == E
    float v = H[b * EE + tid];
    float n2 = breduce_sum<128>(v * v, red);
    HN[b * EE + tid] = v / (sqrtf(n2) + EPSF);
}

__global__ void topk_kernel(const float* __restrict__ SIMM, const float* __restrict__ H,
                            float* __restrict__ HF)
{
    __shared__ float sv[BB];
    __shared__ float beta[10];
    __shared__ int   seli[10];
    const int b = blockIdx.x;
    const int tid = threadIdx.x;       // 128
    for (int j = tid; j < BB; j += 128) sv[j] = SIMM[(long)b * BB + j];
    __syncthreads();
    if (tid == 0) {
        float vals[10];
        for (int k = 0; k < 10; ++k) {
            float best = -1e30f; int bi = 0;
            for (int j = 0; j < BB; ++j)
                if (sv[j] > best) { best = sv[j]; bi = j; }
            vals[k] = best; seli[k] = bi; sv[bi] = -1e30f;
        }
        float mx = vals[0], ssum = 0.f, e[10];
        for (int k = 0; k < 10; ++k) { e[k] = expf(12.f * (vals[k] - mx)); ssum += e[k]; }
        for (int k = 0; k < 10; ++k) beta[k] = e[k] / ssum;
    }
    __syncthreads();
    float hs = 0.f;
    for (int k = 0; k < 10; ++k) hs += beta[k] * H[(long)seli[k] * EE + tid];
    HF[b * EE + tid] = H[b * EE + tid] + 0.3f * hs;    // OMEGA
}

__global__ void cscale_kernel(const float* __restrict__ emb, float* __restrict__ CS)
{
    __shared__ float red[128];
    const int v = blockIdx.x;
    const int tid = threadIdx.x;       // 128 == E
    float e = emb[(long)v * EE + tid];
    float n2 = breduce_sum<128>(e * e, red);
    if (tid == 0) CS[v] = 20.f / (sqrtf(n2) + EPSF);   // W_SCALE / (||emb_v|| + EPS)
}

// ---------------------------------------------------------------------------
extern "C" void kernel_launch(void* const* d_in, const int* in_sizes, int n_in,
                              void* d_out, int out_size, void* d_ws, size_t ws_size,
                              hipStream_t stream)
{
    (void)in_sizes; (void)n_in; (void)out_size; (void)ws_size;
    const int*   alias   = (const int*)  d_in[0];
    const float* Aadj    = (const float*)d_in[1];
    const int*   item    = (const int*)  d_in[2];
    const int*   mask    = (const int*)  d_in[3];
    const float* emb     = (const float*)d_in[4];
    const float* pos     = (const float*)d_in[5];
    const float* W_in    = (const float*)d_in[6];
    const float* W_out   = (const float*)d_in[7];
    const float* b_in    = (const float*)d_in[8];
    const float* b_out   = (const float*)d_in[9];
    const float* W_ih    = (const float*)d_in[10];
    const float* W_hh    = (const float*)d_in[11];
    const float* b_ih    = (const float*)d_in[12];
    const float* b_hh    = (const float*)d_in[13];
    const float* Wa      = (const float*)d_in[14];
    const float* ba      = (const float*)d_in[15];
    const float* Wq      = (const float*)d_in[16];
    const float* bq      = (const float*)d_in[17];
    const float* W1      = (const float*)d_in[18];
    const float* b1      = (const float*)d_in[19];
    const float* W2      = (const float*)d_in[20];
    const float* b2      = (const float*)d_in[21];
    const float* Wc      = (const float*)d_in[22];
    const float* Wg1     = (const float*)d_in[23];
    const float* Wg2     = (const float*)d_in[24];
    const float* Wg0     = (const float*)d_in[25];
    const float* b_alpha = (const float*)d_in[26];
    const float* Wh      = (const float*)d_in[27];
    const float* bh      = (const float*)d_in[28];
    const float* ln_g    = (const float*)d_in[29];
    const float* ln_b    = (const float*)d_in[30];
    float* out = (float*)d_out;
    float* ws  = (float*)d_ws;

    // ---- workspace layout (floats), with phase-based region reuse ----
    size_t o = 0;
    float* X   = ws + o; o += (size_t)BB * LL * DD;        //  6.55M
    float* XTP = ws + o;                                   // 13.11M (also T1|T2)
    float* T1  = XTP;
    float* T2  = XTP + (size_t)BB * LL * DD; o += 2ull * BB * LL * DD;
    float* ADJ = ws + o; o += (size_t)BB * LL * 2 * DD;    // 13.11M (later: Q, PG)
    float* GI  = ws + o; o += (size_t)BB * LL * 3 * DD;    // 19.66M (later: ATT,P,F2)
    float* GH  = ws + o; o += (size_t)BB * LL * 3 * DD;    // 19.66M (later: R)
    float* XN  = ws + o; o += (size_t)BB * LL * DD;        //  6.55M (later: G1)
    float* F1  = ws + o; o += (size_t)BB * L2 * DD;        // 13.11M
    float* CH  = ws + o; o += (size_t)BB * L2 * DD;        // 13.11M
    float* CLAST = ws + o; o += (size_t)BB * DD;
    float* TGT   = ws + o; o += (size_t)BB * DD;
    float* TC    = ws + o; o += (size_t)BB * 2 * DD;
    float* POSV  = ws + o; o += (size_t)BB * DD;
    float* TV    = ws + o; o += (size_t)BB * DD;
    float* SIMV  = ws + o; o += BB;
    float* THR   = ws + o; o += 2;
    float* A1    = ws + o; o += BB;
    float* A2    = ws + o; o += BB;
    float* HCAT  = ws + o; o += (size_t)BB * 2 * DD;
    float* H     = ws + o; o += (size_t)BB * EE;
    float* HN    = ws + o; o += (size_t)BB * EE;
    float* SIMM  = ws + o; o += (size_t)BB * BB;
    float* HF    = ws + o; o += (size_t)BB * EE;
    float* CS    = ws + o; o += VV;
    // region reuse (non-overlapping in time):
    float* Q   = ADJ;                          // after ADJ consumed by GI
    float* ATT = GI;                           // after GRU
    float* P   = GI + (size_t)BB * L2 * L2;    // fits: 2*5.12M < 19.66M
    float* R   = GH;                           // after GRU
    float* F2  = GI;                           // after ATT/P consumed
    float* G1  = XN;                           // after session gather
    float* LOG = X;                            // after GRU
    float* PG  = ADJ;                          // after Q consumed

    const int TPB = 256;
    const float inv_sqrtD = 0.0625f;           // 1/sqrt(256)

    // 1) X = concat(emb[item], pos)
    {
        int total = BB * LL * DD;
        build_x_kernel<<<(total + TPB - 1) / TPB, TPB, 0, stream>>>(emb, pos, item, X, total);
    }
    // 2-3) T1 = X @ W_in^T, T2 = X @ W_out^T
    launch_gemm(stream, 1, 0, X, W_in,  nullptr, nullptr, T1, BB*LL, DD, DD, DD, DD, DD, 0,0,0,0, 1.f, 1);
    launch_gemm(stream, 1, 0, X, W_out, nullptr, nullptr, T2, BB*LL, DD, DD, DD, DD, DD, 0,0,0,0, 1.f, 1);
    // 4-5) adj_in/out = A_{in,out} @ T{1,2} + b  -> ADJ[:, :, 0:D | D:2D]
    launch_gemm(stream, 0, 0, Aadj,      T1, b_in,  nullptr, ADJ,      LL, DD, LL, 2*LL, DD, 2*DD,
                (long)LL*2*LL, (long)LL*DD, (long)LL*2*DD, 0, 1.f, BB);
    launch_gemm(stream, 0, 0, Aadj + LL, T2, b_out, nullptr, ADJ + DD, LL, DD, LL, 2*LL, DD, 2*DD,
                (long)LL*2*LL, (long)LL*DD, (long)LL*2*DD, 0, 1.f, BB);
    // 6) GI = ADJ @ W_ih^T + b_ih   (K=2D, N=3D)
    launch_gemm(stream, 1, 0, ADJ, W_ih, b_ih, nullptr, GI, BB*LL, 3*DD, 2*DD, 2*DD, 2*DD, 3*DD, 0,0,0,0, 1.f, 1);
    // 7) GH = X @ W_hh^T + b_hh     (K=D, N=3D)
    launch_gemm(stream, 1, 0, X, W_hh, b_hh, nullptr, GH, BB*LL, 3*DD, DD, DD, DD, 3*DD, 0,0,0,0, 1.f, 1);
    // 8) GRU -> XN
    {
        int total = BB * LL * DD;
        gru_kernel<<<(total + TPB - 1) / TPB, TPB, 0, stream>>>(GI, GH, X, XN, total);
    }
    // 9) session gather + zero pad -> XTP (B, 2L, D)
    {
        int total = BB * L2 * DD;
        gather_kernel<<<(total + TPB - 1) / TPB, TPB, 0, stream>>>(XN, alias, XTP, total);
    }
    // 10) alpha1
    alpha1_kernel<<<(BB + TPB - 1) / TPB, TPB, 0, stream>>>(ba, A1);
    // 11) Q = relu(XTP @ Wq^T + bq)
    launch_gemm(stream, 1, 1, XTP, Wq, bq, nullptr, Q, BB*L2, DD, DD, DD, DD, DD, 0,0,0,0, 1.f, 1);
    // 12) ATT = Q @ XTP^T / sqrt(D)   (batched, M=N=2L, K=D)
    launch_gemm(stream, 1, 0, Q, XTP, nullptr, nullptr, ATT, L2, L2, DD, DD, DD, L2,
                (long)L2*DD, (long)L2*DD, (long)L2*L2, 0, inv_sqrtD, BB);
    // 13) P = entmax(ATT, alpha1[b])
    entmax_kernel<<<BB * L2, 256, 0, stream>>>(ATT, P, L2, A1, L2);
    // 14) R = P @ XTP  (batched, M=2L, N=D, K=2L)
    launch_gemm(stream, 0, 0, P, XTP, nullptr, nullptr, R, L2, DD, L2, L2, DD, DD,
                (long)L2*L2, (long)L2*DD, (long)L2*DD, 0, 1.f, BB);
    // 15) F1 = relu(R @ W1^T + b1)
    launch_gemm(stream, 1, 1, R, W1, b1, nullptr, F1, BB*L2, DD, DD, DD, DD, DD, 0,0,0,0, 1.f, 1);
    // 16) F2 = F1 @ W2^T + b2
    launch_gemm(stream, 1, 0, F1, W2, b2, nullptr, F2, BB*L2, DD, DD, DD, DD, DD, 0,0,0,0, 1.f, 1);
    // 17) CH = LayerNorm(F2 + R) * g + b
    ln_kernel<<<BB * L2, 256, 0, stream>>>(F2, R, ln_g, ln_b, CH);
    // 18) C_last / target / concat
    clast_kernel<<<BB, 256, 0, stream>>>(CH, mask, CLAST, TGT, TC);
    // 19-20) sim, thr
    sim_kernel<<<BB, 256, 0, stream>>>(CLAST, TGT, SIMV);
    thr_kernel<<<1, 512, 0, stream>>>(SIMV, THR);
    // 21) positive = concat(target, C_last) @ Wc^T
    launch_gemm(stream, 1, 0, TC, Wc, nullptr, nullptr, POSV, BB, DD, 2*DD, 2*DD, 2*DD, DD, 0,0,0,0, 1.f, 1);
    // 22) target = where(sim - thr > 0, positive, target)
    select_kernel<<<(BB * DD + TPB - 1) / TPB, TPB, 0, stream>>>(SIMV, THR, POSV, TGT, BB * DD);
    // 23) alpha2 = sigmoid(target . Wa + ba) + 1
    alpha2_kernel<<<BB, 256, 0, stream>>>(TGT, Wa, ba, A2);
    // 24) TV = target @ Wg2^T + b_alpha
    launch_gemm(stream, 1, 0, TGT, Wg2, b_alpha, nullptr, TV, BB, DD, DD, DD, DD, DD, 0,0,0,0, 1.f, 1);
    // 25) G1 = relu(C @ Wg1^T + TV[b])  (batched; per-batch row bias via sBias)
    launch_gemm(stream, 1, 1, CH, Wg1, TV, nullptr, G1, LL, DD, DD, DD, DD, DD,
                (long)L2*DD, 0, (long)LL*DD, DD, 1.f, BB);
    // 26) LOG = G1 @ Wg0^T
    launch_gemm(stream, 1, 0, G1, Wg0, nullptr, nullptr, LOG, BB*LL, DD, DD, DD, DD, DD, 0,0,0,0, 1.f, 1);
    // 27) PG = entmax(LOG, alpha2[b])
    entmax_kernel<<<BB * LL, 256, 0, stream>>>(LOG, PG, DD, A2, LL);
    // 28) C_global einsum + concat target -> HCAT
    cg_kernel<<<BB, 256, 0, stream>>>(PG, XTP, TGT, HCAT);
    // 29) H = selu(HCAT @ Wh^T + bh)
    launch_gemm(stream, 1, 2, HCAT, Wh, bh, nullptr, H, BB, EE, 2*DD, 2*DD, 2*DD, EE, 0,0,0,0, 1.f, 1);
    // 30) HN = normalize(H)
    hn_kernel<<<BB, 128, 0, stream>>>(H, HN);
    // 31) SIMM = HN @ HN^T
    launch_gemm(stream, 1, 0, HN, HN, nullptr, nullptr, SIMM, BB, BB, EE, EE, EE, BB, 0,0,0,0, 1.f, 1);
    // 32) top-10 + softmax mix -> HF
    topk_kernel<<<BB, 128, 0, stream>>>(SIMM, H, HF);
    // 33) per-row embedding scales: W_SCALE / (||emb_v|| + EPS)
    cscale_kernel<<<VV, 128, 0, stream>>>(emb, CS);
    // 34) out = (HF @ emb^T) * CS[v]
    launch_gemm(stream, 1, 0, HF, emb, nullptr, CS, out, BB, VV, EE, EE, EE, VV, 0,0,0,0, 1.f, 1);
}